// SoftmaxAUCHLoss_11089605558399
// MI455X (gfx1250) — compile-verified
//
#include <hip/hip_runtime.h>
#include <stdint.h>

// ---------------------------------------------------------------------------
// SoftmaxAUCHLoss for MI455X (gfx1250)
//   output: [B,C] f32 logits, labels: [B,C] int {0,1}
//   -> (ce + 0.1*sum(penalty)/15, 0.1*penalty[C-1])
// Memory-bound streaming reduction: 125.8 MB in, 8 B out  (~5.4 us @ 23.3 TB/s).
// Data movement via CDNA5 Tensor Data Mover (TENSOR_LOAD_TO_LDS), double
// buffered, synced with s_wait_tensorcnt.  LDS row reads are bank-conflict
// free (gcd(15,64)==1).  Deterministic two-stage reduction (wave32 shuffles ->
// per-block partials -> single-block finalize), no float atomics.
// ---------------------------------------------------------------------------

#define C_CLS 15
#define BROWS 1048576
#define TILE_ROWS 256
#define TILE_ELTS (TILE_ROWS * C_CLS)   // 3840 elements = 15360 bytes
#define NBLOCKS 1024
#define TILES_PER_BLOCK 4               // 1024 * 4 * 256 = 1048576 rows exactly
#define NPART 46                        // 15 posSum + 15 negSum + 15 posCnt + 1 ce

#if defined(__gfx1250__) && __has_builtin(__builtin_amdgcn_tensor_load_to_lds)
#define USE_TDM 1
#else
#define USE_TDM 0
#endif

#if USE_TDM
#if __has_include(<hip/amd_detail/amd_gfx1250_TDM.h>)
#define TDM_ARGS6 1     // amdgpu-toolchain / therock headers -> 6-arg builtin
#else
#define TDM_ARGS6 0     // ROCm 7.2 clang-22 -> 5-arg builtin
#endif

typedef unsigned int tdm_v4u __attribute__((ext_vector_type(4)));
typedef int          tdm_v4i __attribute__((ext_vector_type(4)));
typedef int          tdm_v8i __attribute__((ext_vector_type(8)));

// Issue one TDM DMA: contiguous 1-D tile of TILE_ELTS 4-byte elements,
// global -> LDS.  Descriptor per CDNA5 ISA.
__device__ __forceinline__ void tdm_tile_load(const void* gsrc, const void* lds_dst)
{
  uint64_t ga = (uint64_t)(uintptr_t)gsrc;
  uint32_t lo = (uint32_t)(uintptr_t)lds_dst;  // low 32b of generic LDS ptr = LDS byte addr

  tdm_v4u g0;
  g0.x = 1u;                                       // count=1, user D#, no gather
  g0.y = lo;                                       // lds_addr
  g0.z = (uint32_t)ga;                             // global_addr[31:0]
  g0.w = (uint32_t)((ga >> 32) & 0x01ffffffu)      // global_addr[56:32]
       | 0x80000000u;                              // type = 2 ("image") at [127:126]

  const uint32_t td0 = (uint32_t)BROWS * (uint32_t)C_CLS;  // tensor_dim0 (elements)
  tdm_v8i g1;
  g1.s0 = (int)0x00020000u;                              // data_size = 2 -> 4 bytes
  g1.s1 = (int)((td0 & 0xffffu) << 16);                  // tensor_dim0[15:0]
  g1.s2 = (int)(((td0 >> 16) & 0xffffu) | (1u << 16));   // tensor_dim0[31:16], tensor_dim1=1
  g1.s3 = (int)(((uint32_t)TILE_ELTS & 0xffffu) << 16);  // tile_dim0 = 3840 (1-D tile)
  g1.s4 = 0;                                             // tile_dim1 = tile_dim2 = 0
  g1.s5 = (int)td0;                                      // tensor_dim0_stride (unused, 1-D)
  g1.s6 = 0;
  g1.s7 = 0;

  tdm_v4i gz = {0, 0, 0, 0};
#if TDM_ARGS6
  tdm_v8i gz8 = {0, 0, 0, 0, 0, 0, 0, 0};
  __builtin_amdgcn_tensor_load_to_lds(g0, g1, gz, gz, gz8, 0);
#else
  __builtin_amdgcn_tensor_load_to_lds(g0, g1, gz, gz, 0);
#endif
}
#endif // USE_TDM

__device__ __forceinline__ float wave_sum(float x)
{
#pragma unroll
  for (int off = 16; off > 0; off >>= 1) x += __shfl_down(x, off, 32);
  return x;
}

// One row: row-softmax CE at first label==1 (argmax of {0,1} row), plus
// per-class sigmoid pos/neg accumulation.  All arrays compile-time indexed.
__device__ __forceinline__ void process_row(const float* vp, const int* lp,
                                            float* posS, float* negS, float* posN,
                                            float& ce)
{
  float v[C_CLS];
  int   l[C_CLS];
#pragma unroll
  for (int c = 0; c < C_CLS; ++c) { v[c] = vp[c]; l[c] = lp[c]; }

  float m = v[0];
#pragma unroll
  for (int c = 1; c < C_CLS; ++c) m = fmaxf(m, v[c]);
  float se = 0.f;
#pragma unroll
  for (int c = 0; c < C_CLS; ++c) se += __expf(v[c] - m);

  float varg = v[0];
  int found = 0;
#pragma unroll
  for (int c = 0; c < C_CLS; ++c) {
    if (!found && (l[c] == 1)) { varg = v[c]; found = 1; }
  }
  ce += m + __logf(se) - varg;        // -log_softmax(v)[argmax(lab)]

#pragma unroll
  for (int c = 0; c < C_CLS; ++c) {
    float s = 1.f / (1.f + __expf(-v[c]));
    float isPos = (l[c] == 1) ? 1.f : 0.f;
    posS[c] += isPos * s;
    posN[c] += isPos;
    negS[c] += (1.f - isPos) * s;
  }
}

__global__ void __launch_bounds__(256)
auc_main_kernel(const float* __restrict__ gOut, const int* __restrict__ gLab,
                float* __restrict__ part)
{
#if USE_TDM
  __shared__ float sLog[2][TILE_ELTS];
  __shared__ int   sLab[2][TILE_ELTS];
#endif
  __shared__ float sRed[8][NPART];

  const int tid  = threadIdx.x;
  const int wid  = tid >> 5;
  const int lane = tid & 31;
  const size_t tileBase = (size_t)blockIdx.x * TILES_PER_BLOCK;

  float posS[C_CLS], negS[C_CLS], posN[C_CLS];
#pragma unroll
  for (int c = 0; c < C_CLS; ++c) { posS[c] = 0.f; negS[c] = 0.f; posN[c] = 0.f; }
  float ce = 0.f;

#if USE_TDM
  // Prologue: wave 0 DMAs tile 0 into buffer 0.
  if (wid == 0) {
    tdm_tile_load(gOut + tileBase * TILE_ELTS, &sLog[0][0]);
    tdm_tile_load(gLab + tileBase * TILE_ELTS, &sLab[0][0]);
  }
  int p = 0;
  for (int k = 0; k < TILES_PER_BLOCK; ++k) {
    if (wid == 0) {
      if (k + 1 < TILES_PER_BLOCK) {
        // Prefetch next tile into the other buffer, then wait for current:
        // in-order TENSORcnt, 2 descriptors per tile -> <=2 means tile k done.
        tdm_tile_load(gOut + (tileBase + k + 1) * TILE_ELTS, &sLog[p ^ 1][0]);
        tdm_tile_load(gLab + (tileBase + k + 1) * TILE_ELTS, &sLab[p ^ 1][0]);
        __builtin_amdgcn_s_wait_tensorcnt(2);
      } else {
        __builtin_amdgcn_s_wait_tensorcnt(0);
      }
    }
    __syncthreads();
    // Lane i, element c -> LDS bank (15*i + c) % 64: conflict free.
    process_row(&sLog[p][tid * C_CLS], &sLab[p][tid * C_CLS], posS, negS, posN, ce);
    __syncthreads();   // all reads done before buffer p is re-DMAed next iter
    p ^= 1;
  }
#else
  for (int k = 0; k < TILES_PER_BLOCK; ++k) {
    size_t row = (tileBase + k) * TILE_ROWS + tid;
    process_row(gOut + row * C_CLS, gLab + row * C_CLS, posS, negS, posN, ce);
  }
#endif

  // Deterministic wave32 shuffle reduction, then per-wave partials to LDS.
#pragma unroll
  for (int c = 0; c < C_CLS; ++c) {
    posS[c] = wave_sum(posS[c]);
    negS[c] = wave_sum(negS[c]);
    posN[c] = wave_sum(posN[c]);
  }
  ce = wave_sum(ce);
  if (lane == 0) {
#pragma unroll
    for (int c = 0; c < C_CLS; ++c) {
      sRed[wid][c]          = posS[c];
      sRed[wid][15 + c]     = negS[c];
      sRed[wid][30 + c]     = posN[c];
    }
    sRed[wid][45] = ce;
  }
  __syncthreads();
  if (tid < NPART) {
    float s = 0.f;
#pragma unroll
    for (int w = 0; w < 8; ++w) s += sRed[w][tid];
    part[(size_t)blockIdx.x * NPART + tid] = s;
  }
}

__global__ void __launch_bounds__(64)
auc_final_kernel(const float* __restrict__ part, float* __restrict__ out)
{
  __shared__ float fin[NPART];
  const int tid = threadIdx.x;
  if (tid < NPART) {
    float s = 0.f;
    for (int b = 0; b < NBLOCKS; ++b) s += part[(size_t)b * NPART + tid];
    fin[tid] = s;
  }
  __syncthreads();
  if (tid == 0) {
    const float Bf = (float)BROWS;
    float ce = fin[45] / Bf;
    float sum_term = 0.f, pen_last = 0.f;
#pragma unroll
    for (int c = 0; c < C_CLS; ++c) {
      float np = fin[30 + c];
      float nn = Bf - np;                       // labels are exactly {0,1}
      float mp = fin[c]      / fmaxf(np, 1.f);
      float mn = fin[15 + c] / fmaxf(nn, 1.f);
      float pen = (np == 0.f) ? (1.f + mn)
                : (nn == 0.f) ? (1.f - mp)
                              : (1.f - mp + mn);
      sum_term += pen;
      if (c == C_CLS - 1) pen_last = pen;
    }
    out[0] = ce + 0.1f * (sum_term / 15.f);
    out[1] = 0.1f * pen_last;
  }
}

extern "C" void kernel_launch(void* const* d_in, const int* in_sizes, int n_in,
                              void* d_out, int out_size, void* d_ws, size_t ws_size,
                              hipStream_t stream)
{
  const float* gOut = (const float*)d_in[0];
  const int*   gLab = (const int*)d_in[1];
  float* part = (float*)d_ws;        // NBLOCKS * NPART floats = 188416 B
  float* out  = (float*)d_out;       // 2 floats

  auc_main_kernel<<<NBLOCKS, 256, 0, stream>>>(gOut, gLab, part);
  auc_final_kernel<<<1, 64, 0, stream>>>(part, out);
}